// MultiQueryDotProductAttention_73839077753281
// MI455X (gfx1250) — compile-verified
//
#include <hip/hip_runtime.h>
#include <hip/hip_bf16.h>
#include <cstdint>

// MQA decode: B=32, F=4096, H=32, D=128, L=8192, fp32 throughout.
// Memory-bound (~300MB traffic); compute via V_WMMA_F32_16X16X4_F32.
// A/Q tiles staged to LDS via GLOBAL_LOAD_ASYNC_TO_LDS_B128 when available.

typedef float v2f __attribute__((ext_vector_type(2)));
typedef float v8f __attribute__((ext_vector_type(8)));
typedef int   v4i __attribute__((ext_vector_type(4)));

#define BATCH 32
#define FDIM  4096
#define HEADS 32
#define DDIM  128
#define LCTX  8192
#define LC    256            // attention L-chunk
#define NC    (LCTX / LC)    // 32 chunks
#define SLD   260            // padded LDS stride for S/P  (260%64==4 -> conflict-free)
#define QLD   132            // padded LDS stride for Q tile
#define KT    256            // GEMM K-tile staged in LDS
#define ALD   260            // padded LDS stride for A tile

#if defined(__HIP_DEVICE_COMPILE__) &&                                   \
    __has_builtin(__builtin_amdgcn_global_load_async_to_lds_b128) &&     \
    __has_builtin(__builtin_amdgcn_s_wait_asynccnt)
#define HAVE_ASYNC_LDS 1
#else
#define HAVE_ASYNC_LDS 0
#endif

#if HAVE_ASYNC_LDS
typedef __attribute__((address_space(1))) v4i gas_v4i;  // global
typedef __attribute__((address_space(3))) v4i las_v4i;  // LDS
#endif

// 16-byte global -> LDS copy (async when the CDNA5 builtin exists).
__device__ __forceinline__ void cp16_g2l(const float* __restrict__ g, float* l) {
#if HAVE_ASYNC_LDS
  __builtin_amdgcn_global_load_async_to_lds_b128(
      (gas_v4i*)(uintptr_t)g, (las_v4i*)(uintptr_t)l, 0, 0);
#else
  *(float4*)l = *(const float4*)g;
#endif
}
__device__ __forceinline__ void cp_g2l_wait() {
#if HAVE_ASYNC_LDS
  __builtin_amdgcn_s_wait_asynccnt(0);
#endif
}

__device__ __forceinline__ v8f wmma4(v2f a, v2f b, v8f c) {
  // D = A(16x4 f32) * B(4x16 f32) + C(16x16 f32)
  return __builtin_amdgcn_wmma_f32_16x16x4_f32(false, a, false, b, (short)0, c,
                                               false, false);
}

// ---------------------------------------------------------------------------
// GEMM: C[32 x N] = A[32 x KDIM] * Bm[KDIM x N] (row-major f32), LDB/K static
// so B-stream addresses fold into global_load immediate offsets.
// 256 threads = 8 waves; wave owns a 32x16 column tile. grid.x = N/128.
// ---------------------------------------------------------------------------
template <int LDB, int KDIM>
__global__ __launch_bounds__(256) void gemm_a32_wmma(
    const float* __restrict__ A, int lda,
    const float* __restrict__ Bm,
    float* __restrict__ C, int ldc) {
  __shared__ float As[32 * ALD];

  const int tid   = threadIdx.x;
  const int wave  = tid >> 5;
  const int lane  = tid & 31;
  const int lhalf = lane >> 4;     // 0 | 1
  const int lm    = lane & 15;
  const int n0    = blockIdx.x * 128 + wave * 16;

  v8f c0 = {};
  v8f c1 = {};

  const int row = tid >> 3;   // 0..31
  const int c4  = tid & 7;    // 0..7

  for (int k0 = 0; k0 < KDIM; k0 += KT) {
    __syncthreads();
    // stage A[0:32, k0:k0+KT] into padded LDS (async 16B copies)
    {
      const float* src = A + (size_t)row * lda + k0;
      float* dst = As + row * ALD;
#pragma unroll
      for (int i = 0; i < KT / 32; ++i)   // 8 x 16B per thread
        cp16_g2l(src + (c4 + 8 * i) * 4, dst + (c4 + 8 * i) * 4);
      cp_g2l_wait();
    }
    __syncthreads();

    const float* brow = Bm + (size_t)k0 * LDB + n0 + lm;
    if (k0 + KT < KDIM)
      __builtin_prefetch(brow + (size_t)KT * LDB, 0, 0);  // next B stage
#pragma unroll 8
    for (int kk = 0; kk < KT; kk += 4) {
      const int kc = kk + 2 * lhalf;
      v2f a0 = *(const v2f*)(As + lm * ALD + kc);
      v2f a1 = *(const v2f*)(As + (lm + 16) * ALD + kc);
      v2f b;
      b.x = brow[(size_t)kc * LDB];
      b.y = brow[(size_t)(kc + 1) * LDB];
      c0 = wmma4(a0, b, c0);
      c1 = wmma4(a1, b, c1);
    }
  }

#pragma unroll
  for (int i = 0; i < 8; ++i) {
    const int m = i + 8 * lhalf;
    C[(size_t)m * ldc + n0 + lm]        = c0[i];
    C[(size_t)(m + 16) * ldc + n0 + lm] = c1[i];
  }
}

// ---------------------------------------------------------------------------
// RoPE on q (1/sqrt(D) folded in) and k; v copied. One block per batch.
// ---------------------------------------------------------------------------
__global__ __launch_bounds__(256) void rope_kernel(
    const float* __restrict__ qraw, const float* __restrict__ kraw,
    const float* __restrict__ vraw, const int* __restrict__ cidx,
    float* __restrict__ qs, float* __restrict__ knew, float* __restrict__ vnew) {
  const int b   = blockIdx.x;
  const int tid = threadIdx.x;
  const float pos   = (float)cidx[b];
  const float scale = 0.08838834764831845f;  // 1/sqrt(128)

  for (int i = tid; i < HEADS * 64; i += 256) {
    const int h = i >> 6, d = i & 63;
    const float freq = __powf(10000.0f, -(float)d * (1.0f / 64.0f));
    float s, c;
    __sincosf(pos * freq, &s, &c);
    const float x1 = qraw[b * FDIM + h * DDIM + d];
    const float x2 = qraw[b * FDIM + h * DDIM + d + 64];
    qs[b * FDIM + h * DDIM + d]      = (x1 * c - x2 * s) * scale;
    qs[b * FDIM + h * DDIM + d + 64] = (x2 * c + x1 * s) * scale;
  }
  if (tid < 64) {
    const int d = tid;
    const float freq = __powf(10000.0f, -(float)d * (1.0f / 64.0f));
    float s, c;
    __sincosf(pos * freq, &s, &c);
    const float x1 = kraw[b * DDIM + d];
    const float x2 = kraw[b * DDIM + d + 64];
    knew[b * DDIM + d]      = x1 * c - x2 * s;
    knew[b * DDIM + d + 64] = x2 * c + x1 * s;
  }
  if (tid < DDIM) vnew[b * DDIM + tid] = vraw[b * DDIM + tid];
}

// ---------------------------------------------------------------------------
// Flash-decoding attention chunk. Block = (batch b, chunk of LC cache cols).
// S = (q*scale) @ K_cache (WMMA), fix column l==idx with q.k_new (caches are
// read-only; reference ADDs new k/v into the slot), mask, partial softmax,
// P@V (WMMA) with v_new correction. Chunks beyond cache_index early-exit.
// ---------------------------------------------------------------------------
__global__ __launch_bounds__(256) void attn_chunk_kernel(
    const float* __restrict__ qs, const float* __restrict__ knew,
    const float* __restrict__ vnew, const float* __restrict__ Kc,
    const float* __restrict__ Vc, const int* __restrict__ cidx,
    float* __restrict__ pm, float* __restrict__ ps, float* __restrict__ pacc) {
  __shared__ float Q[HEADS * QLD];   // q[h][d]
  __shared__ float S[HEADS * SLD];   // logits -> probs [h][l]

  const int tid   = threadIdx.x;
  const int b     = blockIdx.x / NC;
  const int chunk = blockIdx.x % NC;
  const int l0    = chunk * LC;
  const int idx   = cidx[b];
  const int wave  = tid >> 5;
  const int lane  = tid & 31;
  const int lhalf = lane >> 4;
  const int lm    = lane & 15;

  float* pm_b = pm + (b * NC + chunk) * HEADS;
  float* ps_b = ps + (b * NC + chunk) * HEADS;
  float* pa_b = pacc + (size_t)(b * NC + chunk) * (HEADS * DDIM);

  if (l0 > idx) {  // fully masked chunk
    if (tid < HEADS) { pm_b[tid] = -1e30f; ps_b[tid] = 0.0f; }
    for (int i = tid; i < HEADS * DDIM; i += 256) pa_b[i] = 0.0f;
    return;
  }

  // stage q[b] into LDS (async): 4096 floats = 1024 x 16B
  {
    const float* src = qs + b * FDIM;
#pragma unroll
    for (int p = 0; p < 4; ++p) {
      const int j = tid + 256 * p;           // float4 index
      cp16_g2l(src + j * 4, Q + (j >> 5) * QLD + (j & 31) * 4);
    }
    cp_g2l_wait();
  }
  __syncthreads();

  // ---- GEMM1: S[h,l] = sum_d Q[h,d] * K[b,d,l0+l] ----
  const float* Kb = Kc + (size_t)b * DDIM * LCTX;
  for (int t = 0; t < 4; ++t) {                // 32 tiles / 8 waves
    const int ti = wave * 4 + t;
    const int h0 = (ti & 1) * 16;
    const int n0 = (ti >> 1) * 16;
    v8f c = {};
    const float* kcol = Kb + l0 + n0 + lm;
#pragma unroll 8
    for (int kk = 0; kk < DDIM; kk += 4) {
      const int kc = kk + 2 * lhalf;
      v2f a = *(const v2f*)(Q + (h0 + lm) * QLD + kc);
      v2f bb;
      bb.x = kcol[(size_t)kc * LCTX];
      bb.y = kcol[(size_t)(kc + 1) * LCTX];
      c = wmma4(a, bb, c);
    }
#pragma unroll
    for (int i = 0; i < 8; ++i)
      S[(h0 + i + 8 * lhalf) * SLD + n0 + lm] = c[i];
  }
  __syncthreads();

  // ---- fix the l==idx column: key there is cached + k_new ----
  if (idx >= l0 && idx < l0 + LC && tid < HEADS) {
    const float* kn = knew + b * DDIM;
    float dotv = 0.0f;
    for (int d = 0; d < DDIM; ++d) dotv += Q[tid * QLD + d] * kn[d];
    S[tid * SLD + (idx - l0)] += dotv;
  }
  __syncthreads();

  // ---- partial softmax: each wave owns 4 head rows ----
  int cmax = idx - l0;
  if (cmax > LC - 1) cmax = LC - 1;
  for (int j = 0; j < 4; ++j) {
    const int h = wave * 4 + j;
    float mx = -1e30f;
    for (int c = lane; c <= cmax; c += 32) mx = fmaxf(mx, S[h * SLD + c]);
    for (int off = 16; off > 0; off >>= 1) mx = fmaxf(mx, __shfl_xor(mx, off));
    float sum = 0.0f;
    for (int c = lane; c < LC; c += 32) {
      const float p = (c <= cmax) ? __expf(S[h * SLD + c] - mx) : 0.0f;
      S[h * SLD + c] = p;
      sum += p;
    }
    for (int off = 16; off > 0; off >>= 1) sum += __shfl_xor(sum, off);
    if (lane == 0) { pm_b[h] = mx; ps_b[h] = sum; }
  }
  __syncthreads();

  // ---- GEMM2: acc[d,h] = sum_l V[b,d,l0+l] * P[h,l] ----
  const float* Vb = Vc + (size_t)b * DDIM * LCTX;
  const int d0 = wave * 16;                    // wave owns 16 d-rows
  v8f c0 = {};
  v8f c1 = {};
  const float* vrow = Vb + (size_t)(d0 + lm) * LCTX + l0;
#pragma unroll 8
  for (int kk = 0; kk < LC; kk += 4) {
    const int kc = kk + 2 * lhalf;
    v2f a = *(const v2f*)(vrow + kc);
    v2f b0 = *(const v2f*)(S + lm * SLD + kc);         // heads 0..15
    v2f b1 = *(const v2f*)(S + (16 + lm) * SLD + kc);  // heads 16..31
    c0 = wmma4(a, b0, c0);
    c1 = wmma4(a, b1, c1);
  }

  // value at l==idx is cached + v_new -> add v_new * P[h,idx]
  if (idx >= l0 && idx < l0 + LC) {
    const int ic = idx - l0;
    const float p0 = S[lm * SLD + ic];
    const float p1 = S[(16 + lm) * SLD + ic];
#pragma unroll
    for (int i = 0; i < 8; ++i) {
      const float vn = vnew[b * DDIM + d0 + i + 8 * lhalf];
      c0[i] += vn * p0;
      c1[i] += vn * p1;
    }
  }

#pragma unroll
  for (int i = 0; i < 8; ++i) {
    const int d = d0 + i + 8 * lhalf;
    pa_b[lm * DDIM + d]        = c0[i];   // h = lm
    pa_b[(16 + lm) * DDIM + d] = c1[i];   // h = 16+lm
  }
}

// ---------------------------------------------------------------------------
// Combine chunk partials (log-sum-exp) -> attn[b][h*128+d]. Block per (b,h).
// ---------------------------------------------------------------------------
__global__ __launch_bounds__(128) void combine_kernel(
    const float* __restrict__ pm, const float* __restrict__ ps,
    const float* __restrict__ pacc, float* __restrict__ attn) {
  __shared__ float sm[NC], ss[NC];
  const int b = blockIdx.x >> 5;
  const int h = blockIdx.x & 31;
  const int d = threadIdx.x;
  if (d < NC) {
    sm[d] = pm[(b * NC + d) * HEADS + h];
    ss[d] = ps[(b * NC + d) * HEADS + h];
  }
  __syncthreads();
  float M = -1e30f;
  for (int c = 0; c < NC; ++c) M = fmaxf(M, sm[c]);
  float T = 0.0f;
  for (int c = 0; c < NC; ++c) T += ss[c] * __expf(sm[c] - M);
  float acc = 0.0f;
  for (int c = 0; c < NC; ++c) {
    const float w = __expf(sm[c] - M);
    if (w > 0.0f)
      acc += pacc[((size_t)(b * NC + c) * HEADS + h) * DDIM + d] * w;
  }
  attn[b * FDIM + h * DDIM + d] = acc / T;
}

// ---------------------------------------------------------------------------
extern "C" void kernel_launch(void* const* d_in, const int* in_sizes, int n_in,
                              void* d_out, int out_size, void* d_ws,
                              size_t ws_size, hipStream_t stream) {
  (void)in_sizes; (void)n_in; (void)out_size; (void)ws_size;
  const float* x   = (const float*)d_in[0];
  const float* Kc  = (const float*)d_in[1];
  const float* Vc  = (const float*)d_in[2];
  const float* Wq  = (const float*)d_in[3];
  const float* Wk  = (const float*)d_in[4];
  const float* Wv  = (const float*)d_in[5];
  const float* Wo  = (const float*)d_in[6];
  const int*   ci  = (const int*)d_in[7];
  float* out = (float*)d_out;

  float* ws   = (float*)d_ws;
  float* qraw = ws;                       // 32*4096
  float* kraw = qraw + BATCH * FDIM;      // 32*128
  float* vraw = kraw + BATCH * DDIM;      // 32*128
  float* qsws = vraw + BATCH * DDIM;      // 32*4096
  float* knew = qsws + BATCH * FDIM;      // 32*128
  float* vnew = knew + BATCH * DDIM;      // 32*128
  float* pm   = vnew + BATCH * DDIM;      // 32*NC*32
  float* ps   = pm + BATCH * NC * HEADS;  // 32*NC*32
  float* pacc = ps + BATCH * NC * HEADS;  // 32*NC*4096
  float* attn = pacc + (size_t)BATCH * NC * HEADS * DDIM;  // 32*4096

  // Projections: q = x*Wq, k = x*Wk, v = x*Wv
  gemm_a32_wmma<FDIM, FDIM><<<FDIM / 128, 256, 0, stream>>>(x, FDIM, Wq, qraw, FDIM);
  gemm_a32_wmma<DDIM, FDIM><<<1, 256, 0, stream>>>(x, FDIM, Wk, kraw, DDIM);
  gemm_a32_wmma<DDIM, FDIM><<<1, 256, 0, stream>>>(x, FDIM, Wv, vraw, DDIM);

  rope_kernel<<<BATCH, 256, 0, stream>>>(qraw, kraw, vraw, ci, qsws, knew, vnew);

  attn_chunk_kernel<<<BATCH * NC, 256, 0, stream>>>(qsws, knew, vnew, Kc, Vc, ci,
                                                    pm, ps, pacc);

  combine_kernel<<<BATCH * HEADS, 128, 0, stream>>>(pm, ps, pacc, attn);

  // out = attn * Wo   ([32,4096] x [4096,4096])
  gemm_a32_wmma<FDIM, FDIM><<<FDIM / 128, 256, 0, stream>>>(attn, FDIM, Wo, out, FDIM);
}